// MyWaveletModule_73967926771779
// MI455X (gfx1250) — compile-verified
//
#include <hip/hip_runtime.h>

typedef __attribute__((ext_vector_type(2))) float v2f;
typedef __attribute__((ext_vector_type(8))) float v8f;

// db4 dec_lo taps
#define D0 (-0.010597401784997278f)
#define D1 ( 0.032883011666982945f)
#define D2 ( 0.030841381835986965f)
#define D3 (-0.18703481171888114f)
#define D4 (-0.02798376941698385f)
#define D5 ( 0.6308807679295904f)
#define D6 ( 0.7148465705525415f)
#define D7 ( 0.23037781330885523f)

// Correlation kernels (filters flipped, as in the reference):
//   klo[t] = dec_lo[7-t]
//   khi[t] = dec_hi[7-t] = (-1)^t * dec_lo[t]
// Select-chain (v_cndmask) instead of dynamic array index -> no scratch.
__device__ __forceinline__ float tap(int t, int band) {
  float v = 0.f;
  v = (t == 0) ? (band ? ( D0) : D7) : v;
  v = (t == 1) ? (band ? (-D1) : D6) : v;
  v = (t == 2) ? (band ? ( D2) : D5) : v;
  v = (t == 3) ? (band ? (-D3) : D4) : v;
  v = (t == 4) ? (band ? ( D4) : D3) : v;
  v = (t == 5) ? (band ? (-D5) : D2) : v;
  v = (t == 6) ? (band ? ( D6) : D1) : v;
  v = (t == 7) ? (band ? (-D7) : D0) : v;
  return v;
}

// Banded stride-2 filter matrix entry. The halo in LDS starts at global
// index (2*tile_base - 6), so for local output p and tap t the local input
// index is j = 2p + t  =>  t = j - 2p   (j in [0,37] exactly spans the
// 38-wide halo; pad rows/cols 38-39 get strictly zero coefficients).
// Same value serves as stage-1 B fragment (N=p) and stage-2 A fragment (M=p).
__device__ __forceinline__ float gcoef(int j, int p, int band) {
  int t = j - 2 * p;
  float v = tap(t, band);
  return (t >= 0 && t <= 7) ? v : 0.f;
}

// pywt 'symmetric' boundary, single reflection (pad <= 7 << 512)
__device__ __forceinline__ int reflect512(int i) {
  i = (i < 0) ? (-1 - i) : i;
  i = (i > 511) ? (1023 - i) : i;
  return i;
}

__device__ __forceinline__ v8f wmma_f32(v2f a, v2f b, v8f c) {
  return __builtin_amdgcn_wmma_f32_16x16x4_f32(
      /*neg_a=*/false, a, /*neg_b=*/false, b,
      /*c_mod=*/(short)0, c, /*reuse_a=*/false, /*reuse_b=*/false);
}

// One wave (32 threads) per block; each block produces the 4 db4 bands of one
// 16x16 output tile for one (batch, channel) plane.
__global__ __launch_bounds__(32) void dwt2_db4_wmma(const float* __restrict__ x,
                                                    float* __restrict__ out) {
  __shared__ float ldsX[48 * 40];      // 38x38 halo region, zero-padded
  __shared__ float ldsT[2][40 * 16];   // W-pass result, lo/hi bands

  const int lane = threadIdx.x;        // 0..31 (wave32)
  const int bid  = blockIdx.x;
  const int tw = bid & 15;             // 16 tiles along W2=256
  const int th = (bid >> 4) & 15;      // 16 tiles along H2=256
  const int c  = (bid >> 8) & 31;      // C=32
  const int b  = bid >> 13;            // B=8

  const int hh = lane >> 4;            // K-half selector (lanes 0-15 / 16-31)
  const int lp = lane & 15;            // M or N index within fragment

  // ---- constant filter fragments: 10 K-chunks x {lo,hi}, v2f each ----
  v2f Flo[10], Fhi[10];
#pragma unroll
  for (int kc = 0; kc < 10; ++kc) {
    const int j0 = 4 * kc + 2 * hh;
    Flo[kc].x = gcoef(j0,     lp, 0);
    Flo[kc].y = gcoef(j0 + 1, lp, 0);
    Fhi[kc].x = gcoef(j0,     lp, 1);
    Fhi[kc].y = gcoef(j0 + 1, lp, 1);
  }

  // ---- stage 0: halo load (38x38, symmetric-reflected) into LDS ----
  const float* xp = x + (size_t)(b * 32 + c) * (512 * 512);
  const int baseH = 2 * (th * 16) - 6;
  const int baseW = 2 * (tw * 16) - 6;
  for (int idx = lane; idx < 48 * 40; idx += 32) {
    const int row = idx / 40;
    const int col = idx - row * 40;
    float v = 0.f;  // zero pad: keeps NaN out of the K/M overhang
    if (row < 38 && col < 38) {
      const int gh = reflect512(baseH + row);
      const int gw = reflect512(baseW + col);
      v = xp[gh * 512 + gw];
    }
    ldsX[idx] = v;
  }
  __syncthreads();

  // ---- stage 1: W-pass.  T_band(48x16) = X(48x40) * Gw_band(40x16) ----
#pragma unroll
  for (int m = 0; m < 3; ++m) {
    v8f accLo = {};
    v8f accHi = {};
    const int row = m * 16 + lp;  // A-fragment M index
#pragma unroll
    for (int kc = 0; kc < 10; ++kc) {
      const float* p = &ldsX[row * 40 + 4 * kc + 2 * hh];
      v2f a;
      a.x = p[0];
      a.y = p[1];                 // ds_load_b64 (8B aligned)
      accLo = wmma_f32(a, Flo[kc], accLo);
      accHi = wmma_f32(a, Fhi[kc], accHi);
    }
#pragma unroll
    for (int r = 0; r < 8; ++r) {
      const int trow = m * 16 + 8 * hh + r;  // D layout: M = 8*hh + r
      if (trow < 40) {
        ldsT[0][trow * 16 + lp] = accLo[r];
        ldsT[1][trow * 16 + lp] = accHi[r];
      }
    }
  }
  __syncthreads();

  // ---- stage 2: H-pass.  Out(16x16) = Fh_band(16x40) * T(40x16) ----
  float* outPlane = out + (size_t)b * (128 * 256 * 256) + (size_t)c * (256 * 256);
#pragma unroll
  for (int wb = 0; wb < 2; ++wb) {
    v8f accA = {};  // hband = 0 (lo)
    v8f accB = {};  // hband = 1 (hi)
#pragma unroll
    for (int kc = 0; kc < 10; ++kc) {
      const int j0 = 4 * kc + 2 * hh;
      v2f t;
      t.x = ldsT[wb][j0 * 16 + lp];
      t.y = ldsT[wb][(j0 + 1) * 16 + lp];
      accA = wmma_f32(Flo[kc], t, accA);
      accB = wmma_f32(Fhi[kc], t, accB);
    }
    // out channel = wband*64 + hband*32 + c ; NT stores (write-once data,
    // keep L2 for input halo reuse)
#pragma unroll
    for (int r = 0; r < 8; ++r) {
      const int h2 = th * 16 + 8 * hh + r;
      const size_t o = (size_t)h2 * 256 + (size_t)(tw * 16 + lp);
      __builtin_nontemporal_store(accA[r],
          outPlane + (size_t)(wb * 64) * 65536 + o);
      __builtin_nontemporal_store(accB[r],
          outPlane + (size_t)(wb * 64 + 32) * 65536 + o);
    }
  }
}

extern "C" void kernel_launch(void* const* d_in, const int* in_sizes, int n_in,
                              void* d_out, int out_size, void* d_ws, size_t ws_size,
                              hipStream_t stream) {
  (void)in_sizes; (void)n_in; (void)d_ws; (void)ws_size; (void)out_size;
  const float* x = (const float*)d_in[0];
  float* out = (float*)d_out;
  // 8 batches * 32 channels * 16x16 tiles = 65536 single-wave workgroups
  dwt2_db4_wmma<<<dim3(8 * 32 * 16 * 16), dim3(32), 0, stream>>>(x, out);
}